// GatedMultifactorSelfAttnEnc_31164282700424
// MI455X (gfx1250) — compile-verified
//
#include <hip/hip_runtime.h>

// Problem constants (match reference).
#define B_ 8
#define L_ 1024
#define H_ 512
#define F_ 4

typedef float v2f __attribute__((ext_vector_type(2)));
typedef float v8f __attribute__((ext_vector_type(8)));

// D = A(16x4,f32) * B(4x16,f32) + C(16x16,f32) on the CDNA5 matrix pipe.
static __device__ __forceinline__ v8f wmma_f32(v2f a, v2f b, v8f c) {
    return __builtin_amdgcn_wmma_f32_16x16x4_f32(false, a, false, b, (short)0, c,
                                                 false, false);
}

// A-fragment (16x4): lane r in [0,16) holds row (row_base+r); lanes 0-15 carry
// K = k0,k0+1 and lanes 16-31 carry K = k0+2,k0+3 (pass k = k0 + 2*g).
// Row-major source, contiguous 8-byte load per lane.
static __device__ __forceinline__ v2f ld_a(const float* __restrict__ A, int lda,
                                           int row, int k) {
    return *(const v2f*)(A + (size_t)row * lda + k);
}

// B-fragment (4x16) from a row-major matrix: lane r holds column col; K split
// across lane halves identically to A. Two coalesced scalar loads per lane.
static __device__ __forceinline__ v2f ld_b(const float* __restrict__ Bm, int ldb,
                                           int k, int col) {
    v2f v;
    v.x = Bm[(size_t)k * ldb + col];
    v.y = Bm[(size_t)(k + 1) * ldb + col];
    return v;
}

// ---------------------------------------------------------------------------
// Kernel 1: y = relu(x @ W_proj + b_proj), scattered into per-factor layout
// y_perm[b][f][l][h]  (so the scores GEMM has contiguous K).
// M=8192, K=512, N=2048. 64x64 tile per wave, 4 waves per block.
// ---------------------------------------------------------------------------
__global__ __launch_bounds__(128) void proj_relu_kernel(
    const float* __restrict__ x, const float* __restrict__ Wp,
    const float* __restrict__ bp, float* __restrict__ y_perm) {
    const int lane = threadIdx.x & 31;
    const int wave = threadIdx.x >> 5;
    const int r = lane & 15;
    const int g = lane >> 4;
    const int m0 = (blockIdx.y * 4 + wave) * 64;
    const int n0 = blockIdx.x * 64;

    const v8f z = {0.f, 0.f, 0.f, 0.f, 0.f, 0.f, 0.f, 0.f};
    v8f acc[4][4];
#pragma unroll
    for (int i = 0; i < 4; ++i)
#pragma unroll
        for (int j = 0; j < 4; ++j) acc[i][j] = z;

    for (int k0 = 0; k0 < H_; k0 += 4) {
        const int k = k0 + 2 * g;
        v2f af[4], bf[4];
#pragma unroll
        for (int i = 0; i < 4; ++i) af[i] = ld_a(x, H_, m0 + 16 * i + r, k);
#pragma unroll
        for (int j = 0; j < 4; ++j) bf[j] = ld_b(Wp, H_ * F_, k, n0 + 16 * j + r);
#pragma unroll
        for (int i = 0; i < 4; ++i)
#pragma unroll
            for (int j = 0; j < 4; ++j)
                acc[i][j] = wmma_f32(af[i], bf[j], acc[i][j]);
    }

#pragma unroll
    for (int j = 0; j < 4; ++j) {
        const int n = n0 + 16 * j + r;
        const int f = n & 3;
        const int h = n >> 2;
        const float bias = bp[n];
#pragma unroll
        for (int i = 0; i < 4; ++i) {
#pragma unroll
            for (int v = 0; v < 8; ++v) {
                const int m = m0 + 16 * i + 8 * g + v;  // C layout: M=v(+8 hi half)
                const int b = m >> 10;
                const int l = m & (L_ - 1);
                float c = acc[i][j][v] + bias;
                c = fmaxf(c, 0.0f);
                y_perm[((size_t)(b * F_ + f) * L_ + l) * H_ + h] = c;
            }
        }
    }
}

// ---------------------------------------------------------------------------
// Kernel 2: self_attn[b,l,m] = max_f  Y_f[l,:] . Y_f[m,:]   (K=512 per factor)
// B-fragment of Y^T loads identically to an A-fragment of Y (lane-symmetric
// f32 fragment layout). 32x32 tile per wave so best+acc fit in VGPRs.
// ---------------------------------------------------------------------------
__global__ __launch_bounds__(128) void scores_max_kernel(
    const float* __restrict__ y_perm, float* __restrict__ attn) {
    const int lane = threadIdx.x & 31;
    const int wave = threadIdx.x >> 5;
    const int r = lane & 15;
    const int g = lane >> 4;
    const int b = blockIdx.z;
    const int l0 = (blockIdx.y * 4 + wave) * 32;
    const int m0 = blockIdx.x * 32;

    const float NEG = -3.0e38f;
    const v8f negv = {NEG, NEG, NEG, NEG, NEG, NEG, NEG, NEG};
    const v8f z = {0.f, 0.f, 0.f, 0.f, 0.f, 0.f, 0.f, 0.f};
    v8f best[2][2];
#pragma unroll
    for (int i = 0; i < 2; ++i)
#pragma unroll
        for (int j = 0; j < 2; ++j) best[i][j] = negv;

    for (int f = 0; f < F_; ++f) {
        const float* Y = y_perm + (size_t)(b * F_ + f) * L_ * H_;
        v8f acc[2][2];
#pragma unroll
        for (int i = 0; i < 2; ++i)
#pragma unroll
            for (int j = 0; j < 2; ++j) acc[i][j] = z;

        for (int k0 = 0; k0 < H_; k0 += 4) {
            const int k = k0 + 2 * g;
            v2f af[2], bf[2];
#pragma unroll
            for (int i = 0; i < 2; ++i) af[i] = ld_a(Y, H_, l0 + 16 * i + r, k);
#pragma unroll
            for (int j = 0; j < 2; ++j) bf[j] = ld_a(Y, H_, m0 + 16 * j + r, k);
#pragma unroll
            for (int i = 0; i < 2; ++i)
#pragma unroll
                for (int j = 0; j < 2; ++j)
                    acc[i][j] = wmma_f32(af[i], bf[j], acc[i][j]);
        }
#pragma unroll
        for (int i = 0; i < 2; ++i)
#pragma unroll
            for (int j = 0; j < 2; ++j)
#pragma unroll
                for (int v = 0; v < 8; ++v)
                    best[i][j][v] = fmaxf(best[i][j][v], acc[i][j][v]);
    }

#pragma unroll
    for (int i = 0; i < 2; ++i)
#pragma unroll
        for (int j = 0; j < 2; ++j)
#pragma unroll
            for (int v = 0; v < 8; ++v) {
                const int l = l0 + 16 * i + 8 * g + v;
                const int m = m0 + 16 * j + r;
                attn[((size_t)b * L_ + l) * L_ + m] = best[i][j][v];
            }
}

// ---------------------------------------------------------------------------
// Kernel 3: in-place allennlp-style masked softmax over last axis.
// alpha_j = e_j*m_j / (sum(e*m) + eps*sum(e)),  e_j = exp(t_j - max(t)),
// t_j = logits_j * m_j  (masked entries contribute exp(0) to sum(e), matching
// the reference exactly). One 256-thread block per row (4 elems/thread).
// ---------------------------------------------------------------------------
__global__ __launch_bounds__(256) void masked_softmax_kernel(
    const int* __restrict__ x_mask, float* __restrict__ attn) {
    const int row = blockIdx.x;  // b*L + l
    const int b = row >> 10;
    const int l = row & (L_ - 1);
    float* logits = attn + (size_t)row * L_;
    const int* mrow = x_mask + b * L_;
    const float ml = (float)mrow[l];
    const int tid = threadIdx.x;

    float t[4], mk[4];
    float lmax = -3.0e38f;
#pragma unroll
    for (int u = 0; u < 4; ++u) {
        const int j = tid + 256 * u;
        float mj = ml * (float)mrow[j];
        if (j == l) mj = 0.0f;  // zero diagonal
        const float tv = logits[j] * mj;
        t[u] = tv;
        mk[u] = mj;
        lmax = fmaxf(lmax, tv);
    }

    __shared__ float s1[256];
    __shared__ float s2[256];
    s1[tid] = lmax;
    __syncthreads();
    for (int off = 128; off > 0; off >>= 1) {
        if (tid < off) s1[tid] = fmaxf(s1[tid], s1[tid + off]);
        __syncthreads();
    }
    const float rowmax = s1[0];
    __syncthreads();

    float se = 0.f, spm = 0.f;
#pragma unroll
    for (int u = 0; u < 4; ++u) {
        const float e = __expf(t[u] - rowmax);
        t[u] = e;
        se += e;
        spm += e * mk[u];
    }
    s1[tid] = se;
    s2[tid] = spm;
    __syncthreads();
    for (int off = 128; off > 0; off >>= 1) {
        if (tid < off) {
            s1[tid] += s1[tid + off];
            s2[tid] += s2[tid + off];
        }
        __syncthreads();
    }
    const float inv = 1.0f / (s2[0] + 1e-13f * s1[0]);
#pragma unroll
    for (int u = 0; u < 4; ++u) {
        const int j = tid + 256 * u;
        logits[j] = t[u] * mk[u] * inv;
    }
}

// ---------------------------------------------------------------------------
// Kernel 4: enc[b,l,h] = sum_m alpha[b,l,m] * x[b,m,h].  M=1024,N=512,K=1024.
// ---------------------------------------------------------------------------
__global__ __launch_bounds__(128) void attn_enc_kernel(
    const float* __restrict__ attn, const float* __restrict__ x,
    float* __restrict__ enc) {
    const int lane = threadIdx.x & 31;
    const int wave = threadIdx.x >> 5;
    const int r = lane & 15;
    const int g = lane >> 4;
    const int b = blockIdx.z;
    const float* A = attn + (size_t)b * L_ * L_;
    const float* X = x + (size_t)b * L_ * H_;
    const int l0 = (blockIdx.y * 4 + wave) * 64;
    const int n0 = blockIdx.x * 64;

    const v8f z = {0.f, 0.f, 0.f, 0.f, 0.f, 0.f, 0.f, 0.f};
    v8f acc[4][4];
#pragma unroll
    for (int i = 0; i < 4; ++i)
#pragma unroll
        for (int j = 0; j < 4; ++j) acc[i][j] = z;

    for (int k0 = 0; k0 < L_; k0 += 4) {
        const int k = k0 + 2 * g;
        v2f af[4], bf[4];
#pragma unroll
        for (int i = 0; i < 4; ++i) af[i] = ld_a(A, L_, l0 + 16 * i + r, k);
#pragma unroll
        for (int j = 0; j < 4; ++j) bf[j] = ld_b(X, H_, k, n0 + 16 * j + r);
#pragma unroll
        for (int i = 0; i < 4; ++i)
#pragma unroll
            for (int j = 0; j < 4; ++j)
                acc[i][j] = wmma_f32(af[i], bf[j], acc[i][j]);
    }

#pragma unroll
    for (int i = 0; i < 4; ++i)
#pragma unroll
        for (int j = 0; j < 4; ++j)
#pragma unroll
            for (int v = 0; v < 8; ++v) {
                const int l = l0 + 16 * i + 8 * g + v;
                const int n = n0 + 16 * j + r;
                enc[((size_t)b * L_ + l) * H_ + n] = acc[i][j][v];
            }
}

// ---------------------------------------------------------------------------
// Kernel 5: joint = [x || enc] (not materialized); out = sigmoid(joint@Wg + bg)
// * joint.  M=8192, N=1024, K=1024.  K pairs are even-aligned so a fragment
// never straddles the x/enc boundary at K=512.
// ---------------------------------------------------------------------------
__global__ __launch_bounds__(128) void gate_kernel(
    const float* __restrict__ x, const float* __restrict__ enc,
    const float* __restrict__ Wg, const float* __restrict__ bg,
    float* __restrict__ out) {
    const int lane = threadIdx.x & 31;
    const int wave = threadIdx.x >> 5;
    const int r = lane & 15;
    const int g = lane >> 4;
    const int m0 = (blockIdx.y * 4 + wave) * 64;
    const int n0 = blockIdx.x * 64;

    const v8f z = {0.f, 0.f, 0.f, 0.f, 0.f, 0.f, 0.f, 0.f};
    v8f acc[4][4];
#pragma unroll
    for (int i = 0; i < 4; ++i)
#pragma unroll
        for (int j = 0; j < 4; ++j) acc[i][j] = z;

    for (int k0 = 0; k0 < 2 * H_; k0 += 4) {
        const int k = k0 + 2 * g;
        v2f af[4], bf[4];
#pragma unroll
        for (int i = 0; i < 4; ++i) {
            const int row = m0 + 16 * i + r;
            const float* base = (k < H_) ? (x + (size_t)row * H_ + k)
                                         : (enc + (size_t)row * H_ + (k - H_));
            af[i] = *(const v2f*)base;
        }
#pragma unroll
        for (int j = 0; j < 4; ++j) bf[j] = ld_b(Wg, 2 * H_, k, n0 + 16 * j + r);
#pragma unroll
        for (int i = 0; i < 4; ++i)
#pragma unroll
            for (int j = 0; j < 4; ++j)
                acc[i][j] = wmma_f32(af[i], bf[j], acc[i][j]);
    }

#pragma unroll
    for (int j = 0; j < 4; ++j) {
        const int n = n0 + 16 * j + r;
        const float bias = bg[n];
#pragma unroll
        for (int i = 0; i < 4; ++i) {
#pragma unroll
            for (int v = 0; v < 8; ++v) {
                const int m = m0 + 16 * i + 8 * g + v;
                const float jv = (n < H_) ? x[(size_t)m * H_ + n]
                                          : enc[(size_t)m * H_ + (n - H_)];
                const float s = acc[i][j][v] + bias;
                const float gv = 1.0f / (1.0f + __expf(-s));
                out[(size_t)m * (2 * H_) + n] = gv * jv;
            }
        }
    }
}

// ---------------------------------------------------------------------------
// Host-side launcher. Workspace layout (floats):
//   [0, 16M)   y_perm  : (B,F,L,H)   64 MB
//   [16M, 24M) attn    : (B,L,L)     32 MB  (scores -> softmax in place)
//   [24M, 28M) enc     : (B,L,H)     16 MB
// ---------------------------------------------------------------------------
extern "C" void kernel_launch(void* const* d_in, const int* in_sizes, int n_in,
                              void* d_out, int out_size, void* d_ws,
                              size_t ws_size, hipStream_t stream) {
    (void)in_sizes; (void)n_in; (void)out_size; (void)ws_size;
    const float* x      = (const float*)d_in[0];
    const int*   x_mask = (const int*)d_in[1];
    const float* W_proj = (const float*)d_in[2];
    const float* b_proj = (const float*)d_in[3];
    const float* W_gate = (const float*)d_in[4];
    const float* b_gate = (const float*)d_in[5];
    float* out = (float*)d_out;

    float* ws     = (float*)d_ws;
    float* y_perm = ws;                                   // 16,777,216 floats
    float* attn   = ws + (size_t)B_ * F_ * L_ * H_;       //  8,388,608 floats
    float* enc    = attn + (size_t)B_ * L_ * L_;          //  4,194,304 floats

    // 1) projection + relu, scattered per factor
    proj_relu_kernel<<<dim3((H_ * F_) / 64, (B_ * L_) / 256), 128, 0, stream>>>(
        x, W_proj, b_proj, y_perm);
    // 2) per-factor Gram matrices, max over factors
    scores_max_kernel<<<dim3(L_ / 32, L_ / 128, B_), 128, 0, stream>>>(y_perm,
                                                                       attn);
    // 3) masked softmax (in place)
    masked_softmax_kernel<<<dim3(B_ * L_), 256, 0, stream>>>(x_mask, attn);
    // 4) enc = alpha @ x
    attn_enc_kernel<<<dim3(H_ / 64, L_ / 256, B_), 128, 0, stream>>>(attn, x,
                                                                     enc);
    // 5) gated output
    gate_kernel<<<dim3((2 * H_) / 64, (B_ * L_) / 256), 128, 0, stream>>>(
        x, enc, W_gate, b_gate, out);
}